// PaiNN_38929583571220
// MI455X (gfx1250) — compile-verified
//
#include <hip/hip_runtime.h>
#include <math.h>
#include <stdint.h>

// ---------------- problem constants ----------------
#define NN     10000      // nodes
#define NE     200000     // edges
#define FDIM   128
#define F3     384        // 3*F
#define NRBF   20
#define NLAY   3
#define CUT    5.0f
#define PI_F   3.14159265358979323846f

typedef float v2f __attribute__((ext_vector_type(2)));
typedef float v8f __attribute__((ext_vector_type(8)));

__device__ __forceinline__ float silu_f(float x) {
    return x / (1.0f + expf(-x));
}

// ---------------- utility kernels ----------------
__global__ void k_zero(float* __restrict__ p, int n) {
    int i = blockIdx.x * blockDim.x + threadIdx.x;
    int stride = gridDim.x * blockDim.x;
    for (; i < n; i += stride) p[i] = 0.0f;
}

// s[n,f] = emb[atoms[n], f]
__global__ void k_embed(const int* __restrict__ atoms, const float* __restrict__ emb,
                        float* __restrict__ s) {
    int n = blockIdx.x, f = threadIdx.x;
    s[n * FDIM + f] = emb[atoms[n] * FDIM + f];
}

// per-edge geometry: dist, dir, fcut, rbf
__global__ void k_edge_geom(const int* __restrict__ ei, const float* __restrict__ pos,
                            float* __restrict__ rbf, float* __restrict__ fcut,
                            float* __restrict__ dir) {
    int e = blockIdx.x * blockDim.x + threadIdx.x;
    if (e >= NE) return;
    int dst = ei[2 * e + 0];
    int src = ei[2 * e + 1];
    float rx = pos[src * 3 + 0] - pos[dst * 3 + 0];
    float ry = pos[src * 3 + 1] - pos[dst * 3 + 1];
    float rz = pos[src * 3 + 2] - pos[dst * 3 + 2];
    float d = sqrtf(rx * rx + ry * ry + rz * rz);
    float inv = 1.0f / d;                    // IEEE inf/nan if d==0, matches reference
    dir[e * 3 + 0] = rx * inv;
    dir[e * 3 + 1] = ry * inv;
    dir[e * 3 + 2] = rz * inv;
    fcut[e] = (d < CUT) ? 0.5f * (cosf(PI_F * d / CUT) + 1.0f) : 0.0f;
    float w = d * (PI_F / CUT);
    #pragma unroll
    for (int k = 0; k < NRBF; ++k)
        rbf[e * NRBF + k] = sinf(w * (float)(k + 1)) * inv;
}

// ---------------- fp32 WMMA GEMM:  Y[M,N] = act(X[M,K] @ W[K,N] + b[N]) ------
// One wave per 16x64 output strip (4 accumulators, A fragment reused 4x).
// Requires M % 16 == 0, K % 4 == 0, N % 64 == 0.
// A frag (16x4 f32, 2 VGPR): lane<16 -> A[lane][k0..k0+1]; lane>=16 -> A[lane-16][k0+2..3]
// C/D (16x16 f32, 8 VGPR):   VGPR i -> row m0+i (lanes 0-15) / m0+8+i (lanes 16-31)
__global__ __launch_bounds__(32)
void k_gemm_wmma(const float* __restrict__ X, const float* __restrict__ W,
                 const float* __restrict__ bias, float* __restrict__ Y,
                 int M, int K, int N, int act) {
    int m0 = blockIdx.y * 16;
    int n0 = blockIdx.x * 64;
    int lane = threadIdx.x & 31;
    int lo = lane & 15;
    int hi = lane >> 4;                 // 0 for lanes 0-15, 1 for 16-31
    int col = n0 + lo;

    v8f c0, c1, c2, c3;
    {
        float b0 = bias[col], b1 = bias[col + 16], b2 = bias[col + 32], b3 = bias[col + 48];
        #pragma unroll
        for (int i = 0; i < 8; ++i) { c0[i] = b0; c1[i] = b1; c2[i] = b2; c3[i] = b3; }
    }

    const float* xrow = X + (size_t)(m0 + lo) * K;
    for (int k0 = 0; k0 < K; k0 += 4) {
        int ka = k0 + (hi ? 2 : 0);
        v2f a;
        a[0] = xrow[ka];
        a[1] = xrow[ka + 1];
        const float* wr0 = W + (size_t)ka * N + col;
        const float* wr1 = wr0 + N;
        v2f b0, b1, b2, b3;
        b0[0] = wr0[0];  b0[1] = wr1[0];
        b1[0] = wr0[16]; b1[1] = wr1[16];
        b2[0] = wr0[32]; b2[1] = wr1[32];
        b3[0] = wr0[48]; b3[1] = wr1[48];
        c0 = __builtin_amdgcn_wmma_f32_16x16x4_f32(false, a, false, b0, (short)0, c0, false, false);
        c1 = __builtin_amdgcn_wmma_f32_16x16x4_f32(false, a, false, b1, (short)0, c1, false, false);
        c2 = __builtin_amdgcn_wmma_f32_16x16x4_f32(false, a, false, b2, (short)0, c2, false, false);
        c3 = __builtin_amdgcn_wmma_f32_16x16x4_f32(false, a, false, b3, (short)0, c3, false, false);
    }

    int rbase = m0 + (hi ? 8 : 0);
    #pragma unroll
    for (int i = 0; i < 8; ++i) {
        float y0 = c0[i], y1 = c1[i], y2 = c2[i], y3 = c3[i];
        if (act) { y0 = silu_f(y0); y1 = silu_f(y1); y2 = silu_f(y2); y3 = silu_f(y3); }
        float* yr = Y + (size_t)(rbase + i) * N + col;
        yr[0]  = y0;
        yr[16] = y1;
        yr[32] = y2;
        yr[48] = y3;
    }
}

// ---------------- fused edge message kernel ----------------
// Computes W-row on the fly from rbf (K=20) with rbf_w staged in LDS via
// CDNA5 async global->LDS copies (only the used column chunks [0:128) and
// [256:384)), multiplies by gathered phi[src], forms v/s messages and
// scatters with fp32 atomics into ds/dv.
#define EPB 32   // edges per block (NE % EPB == 0)
__global__ __launch_bounds__(256)
void k_edge_msg(const int* __restrict__ ei, const float* __restrict__ rbf,
                const float* __restrict__ fcut, const float* __restrict__ dir,
                const float* __restrict__ rbfw,   // [20,384] layer slice
                const float* __restrict__ rbfb,   // [384]  layer slice
                const float* __restrict__ phi,    // [NN,384]
                const float* __restrict__ v,      // [NN,3,128]
                float* __restrict__ ds,           // [NN,128]
                float* __restrict__ dv) {         // [NN,3,128]
    __shared__ float lw[NRBF * 256];   // cols: j<128 -> j ; j>=128 -> j+128 (orig 256+f)
    __shared__ float lb[256];

    int tid = threadIdx.x;

    // --- async stage: 20x256 weight slice, 16B per lane per issue ---
    for (int i = tid * 4; i < NRBF * 256; i += 256 * 4) {
        int k = i >> 8;          // /256
        int j = i & 255;         // multiple of 4, never straddles the 128 boundary
        int col = (j < 128) ? j : (j + 128);
        uint32_t laddr = (uint32_t)(uintptr_t)(&lw[i]);
        uint64_t gaddr = (uint64_t)(uintptr_t)(rbfw + (size_t)k * F3 + col);
        asm volatile("global_load_async_to_lds_b128 %0, %1, off"
                     :: "v"(laddr), "v"(gaddr) : "memory");
    }
    if (tid < 64) {
        int j = tid * 4;
        int col = (j < 128) ? j : (j + 128);
        uint32_t laddr = (uint32_t)(uintptr_t)(&lb[j]);
        uint64_t gaddr = (uint64_t)(uintptr_t)(rbfb + col);
        asm volatile("global_load_async_to_lds_b128 %0, %1, off"
                     :: "v"(laddr), "v"(gaddr) : "memory");
    }
    asm volatile("s_wait_asynccnt 0x0" ::: "memory");
    __syncthreads();

    int ebase = blockIdx.x * EPB;
    int f = tid & 127;
    int half = tid >> 7;                 // 0/1 -> which of 2 edges this iter

    for (int it = 0; it < EPB / 2; ++it) {
        int e = ebase + it * 2 + half;
        int dst = ei[2 * e + 0];
        int src = ei[2 * e + 1];

        // prefetch next iteration's gather rows (random-index, L2-resident)
        if (it + 1 < EPB / 2) {
            int en = e + 2;
            int sn = ei[2 * en + 1];
            __builtin_prefetch(&phi[(size_t)sn * F3 + f], 0, 0);
            __builtin_prefetch(&v[(size_t)sn * 3 * FDIM + f], 0, 0);
        }

        float fc = fcut[e];
        const float* re = rbf + (size_t)e * NRBF;

        float wvv = lb[f];
        float wvs = lb[128 + f];
        #pragma unroll
        for (int k = 0; k < NRBF; ++k) {
            float r = re[k];
            wvv = fmaf(r, lw[k * 256 + f], wvv);
            wvs = fmaf(r, lw[k * 256 + 128 + f], wvs);
        }
        wvv *= fc;
        wvs *= fc;

        const float* ps = phi + (size_t)src * F3;
        float mvv = wvv * ps[f];            // split[:, :F]
        float mvs = wvs * ps[256 + f];      // split[:, 2F:]

        atomicAdd(&ds[(size_t)dst * FDIM + f], mvs);

        const float* vs = v + (size_t)src * 3 * FDIM;
        float* dvd = dv + (size_t)dst * 3 * FDIM;
        #pragma unroll
        for (int d = 0; d < 3; ++d) {
            float vm = fmaf(vs[d * FDIM + f], mvv, mvs * dir[e * 3 + d]);
            atomicAdd(&dvd[d * FDIM + f], vm);
        }
    }
}

// s += ds ; v += dv
__global__ void k_apply_delta(float* __restrict__ s, const float* __restrict__ ds,
                              float* __restrict__ v, const float* __restrict__ dv) {
    int i = blockIdx.x * blockDim.x + threadIdx.x;
    if (i < NN * 3 * FDIM) v[i] += dv[i];
    if (i < NN * FDIM)     s[i] += ds[i];
}

// X[n, :128] = ||Vv[n,:,f]|| ;  X[n, 128:256] = s[n,f]
__global__ void k_prep_concat(const float* __restrict__ Vv, const float* __restrict__ s,
                              float* __restrict__ X) {
    int n = blockIdx.x, f = threadIdx.x;
    const float* vn = Vv + (size_t)n * 3 * FDIM;
    float a0 = vn[f], a1 = vn[FDIM + f], a2 = vn[2 * FDIM + f];
    X[(size_t)n * 256 + f] = sqrtf(a0 * a0 + a1 * a1 + a2 * a2);
    X[(size_t)n * 256 + 128 + f] = s[(size_t)n * FDIM + f];
}

// v += a_vv * Uv ; s += a_sv * <Uv,Vv>_d + a_ss
__global__ void k_final_update(float* __restrict__ s, float* __restrict__ v,
                               const float* __restrict__ a, const float* __restrict__ Uv,
                               const float* __restrict__ Vv) {
    int n = blockIdx.x, f = threadIdx.x;
    const float* an = a + (size_t)n * F3;
    float avv = an[f], asv = an[FDIM + f], ass = an[256 + f];
    const float* un = Uv + (size_t)n * 3 * FDIM;
    const float* vn = Vv + (size_t)n * 3 * FDIM;
    float dot = 0.0f;
    #pragma unroll
    for (int d = 0; d < 3; ++d) dot = fmaf(un[d * FDIM + f], vn[d * FDIM + f], dot);
    s[(size_t)n * FDIM + f] += asv * dot + ass;
    float* vd = v + (size_t)n * 3 * FDIM;
    #pragma unroll
    for (int d = 0; d < 3; ++d) vd[d * FDIM + f] += avv * un[d * FDIM + f];
}

// out[n] = h[n,:] . w2[:,0] + b2
__global__ void k_head(const float* __restrict__ h, const float* __restrict__ w2,
                       const float* __restrict__ b2, float* __restrict__ out) {
    int n = blockIdx.x * blockDim.x + threadIdx.x;
    if (n >= NN) return;
    float acc = b2[0];
    const float* hr = h + (size_t)n * FDIM;
    #pragma unroll 8
    for (int i = 0; i < FDIM; ++i) acc = fmaf(hr[i], w2[i], acc);
    out[n] = acc;
}

// ---------------- host orchestration ----------------
extern "C" void kernel_launch(void* const* d_in, const int* in_sizes, int n_in,
                              void* d_out, int out_size, void* d_ws, size_t ws_size,
                              hipStream_t stream) {
    const int*   atoms  = (const int*)d_in[0];
    const float* pos    = (const float*)d_in[1];
    const int*   ei     = (const int*)d_in[2];
    const float* emb    = (const float*)d_in[3];
    const float* phi_w1 = (const float*)d_in[4];
    const float* phi_b1 = (const float*)d_in[5];
    const float* phi_w2 = (const float*)d_in[6];
    const float* phi_b2 = (const float*)d_in[7];
    const float* rbf_w  = (const float*)d_in[8];
    const float* rbf_b  = (const float*)d_in[9];
    const float* u_w    = (const float*)d_in[10];
    const float* u_b    = (const float*)d_in[11];
    const float* v_w    = (const float*)d_in[12];
    const float* v_b    = (const float*)d_in[13];
    const float* up_w1  = (const float*)d_in[14];
    const float* up_b1  = (const float*)d_in[15];
    const float* up_w2  = (const float*)d_in[16];
    const float* up_b2  = (const float*)d_in[17];
    const float* lw1    = (const float*)d_in[18];
    const float* lb1    = (const float*)d_in[19];
    const float* lw2    = (const float*)d_in[20];
    const float* lb2    = (const float*)d_in[21];

    float* ws = (float*)d_ws;
    float* s    = ws;                         // NN*128
    float* v    = s    + (size_t)NN * FDIM;   // NN*3*128
    float* ds   = v    + (size_t)NN * 3 * FDIM;
    float* dv   = ds   + (size_t)NN * FDIM;
    float* phi  = dv   + (size_t)NN * 3 * FDIM;   // NN*384
    float* tmp1 = phi  + (size_t)NN * F3;         // NN*128
    float* Uv   = tmp1 + (size_t)NN * FDIM;       // NN*3*128
    float* Vv   = Uv   + (size_t)NN * 3 * FDIM;
    float* Xc   = Vv   + (size_t)NN * 3 * FDIM;   // NN*256
    float* am   = Xc   + (size_t)NN * 256;        // NN*384
    float* rbf  = am   + (size_t)NN * F3;         // NE*20
    float* fc   = rbf  + (size_t)NE * NRBF;       // NE
    float* dir  = fc   + (size_t)NE;              // NE*3

    float* out = (float*)d_out;

    // init
    k_zero<<<2048, 256, 0, stream>>>(v, NN * 3 * FDIM);
    k_embed<<<NN, FDIM, 0, stream>>>(atoms, emb, s);
    k_edge_geom<<<(NE + 255) / 256, 256, 0, stream>>>(ei, pos, rbf, fc, dir);

    for (int l = 0; l < NLAY; ++l) {
        // phi = silu(s @ phi_w1 + b1) @ phi_w2 + b2
        k_gemm_wmma<<<dim3(FDIM / 64, NN / 16), 32, 0, stream>>>(
            s, phi_w1 + (size_t)l * FDIM * FDIM, phi_b1 + (size_t)l * FDIM,
            tmp1, NN, FDIM, FDIM, 1);
        k_gemm_wmma<<<dim3(F3 / 64, NN / 16), 32, 0, stream>>>(
            tmp1, phi_w2 + (size_t)l * FDIM * F3, phi_b2 + (size_t)l * F3,
            phi, NN, FDIM, F3, 0);

        // messages
        k_zero<<<2048, 256, 0, stream>>>(ds, NN * FDIM);
        k_zero<<<2048, 256, 0, stream>>>(dv, NN * 3 * FDIM);
        k_edge_msg<<<NE / EPB, 256, 0, stream>>>(
            ei, rbf, fc, dir,
            rbf_w + (size_t)l * NRBF * F3, rbf_b + (size_t)l * F3,
            phi, v, ds, dv);
        k_apply_delta<<<(NN * 3 * FDIM + 255) / 256, 256, 0, stream>>>(s, ds, v, dv);

        // Uv / Vv : (3*NN) x 128 @ 128 x 128
        k_gemm_wmma<<<dim3(FDIM / 64, (3 * NN) / 16), 32, 0, stream>>>(
            v, u_w + (size_t)l * FDIM * FDIM, u_b + (size_t)l * FDIM,
            Uv, 3 * NN, FDIM, FDIM, 0);
        k_gemm_wmma<<<dim3(FDIM / 64, (3 * NN) / 16), 32, 0, stream>>>(
            v, v_w + (size_t)l * FDIM * FDIM, v_b + (size_t)l * FDIM,
            Vv, 3 * NN, FDIM, FDIM, 0);

        // a = silu([||Vv||, s] @ up_w1 + b1) @ up_w2 + b2
        k_prep_concat<<<NN, FDIM, 0, stream>>>(Vv, s, Xc);
        k_gemm_wmma<<<dim3(FDIM / 64, NN / 16), 32, 0, stream>>>(
            Xc, up_w1 + (size_t)l * 256 * FDIM, up_b1 + (size_t)l * FDIM,
            tmp1, NN, 256, FDIM, 1);
        k_gemm_wmma<<<dim3(F3 / 64, NN / 16), 32, 0, stream>>>(
            tmp1, up_w2 + (size_t)l * FDIM * F3, up_b2 + (size_t)l * F3,
            am, NN, FDIM, F3, 0);

        k_final_update<<<NN, FDIM, 0, stream>>>(s, v, am, Uv, Vv);
    }

    // head: out = silu(s @ lw1 + lb1) @ lw2 + lb2
    k_gemm_wmma<<<dim3(FDIM / 64, NN / 16), 32, 0, stream>>>(
        s, lw1, lb1, tmp1, NN, FDIM, FDIM, 1);
    k_head<<<(NN + 127) / 128, 128, 0, stream>>>(tmp1, lw2, lb2, out);
    (void)in_sizes; (void)n_in; (void)out_size; (void)ws_size;
}